// Mixture_15685220565523
// MI455X (gfx1250) — compile-verified
//
#include <hip/hip_runtime.h>
#include <hip/hip_bf16.h>
#include <math.h>

typedef __attribute__((ext_vector_type(2))) float v2f;
typedef __attribute__((ext_vector_type(4))) float v4f;
typedef __attribute__((ext_vector_type(8))) float v8f;

#define DDIM 64
#define TWO_PI_F 6.2831853071795864769f
#define EPS_F 1e-8f

// Fragment-order ("swizzled") index for invSigma so that a lane's fp32 WMMA
// B-fragment pair {invS[kr][col], invS[kr+1][col]} is contiguous in memory:
//   row = 4c + 2hi + s, col = 16j + lo
//   idx = (((c*2 + hi) * 4 + j) * 16 + lo) * 2 + s
__device__ __forceinline__ int swz(int row, int col) {
    const int c  = row >> 2;
    const int hi = (row >> 1) & 1;
    const int s  = row & 1;
    const int j  = col >> 4;
    const int lo = col & 15;
    return (((c * 2 + hi) << 6) + (j << 4) + lo) * 2 + s;
}

// ---------------------------------------------------------------------------
// Kernel 1: 64x64 Gauss-Jordan inverse + determinant, one workgroup (64 thr).
// Writes invSigma in WMMA-fragment order (swz) + det.
// ---------------------------------------------------------------------------
__global__ void invdet64_kernel(const float* __restrict__ Sigma,
                                float* __restrict__ invSw,
                                float* __restrict__ detOut) {
    __shared__ float M[DDIM][2 * DDIM + 1];   // +1 pad vs bank conflicts
    const int t = threadIdx.x;                // 0..63

    for (int j = 0; j < DDIM; ++j) {
        M[t][j]        = Sigma[t * DDIM + j];
        M[t][DDIM + j] = (t == j) ? 1.0f : 0.0f;
    }
    __syncthreads();

    float det = 1.0f;
    for (int k = 0; k < DDIM; ++k) {
        // Redundant (identical) partial-pivot search on every thread.
        int   p    = k;
        float best = fabsf(M[k][k]);
        for (int i = k + 1; i < DDIM; ++i) {
            float v = fabsf(M[i][k]);
            if (v > best) { best = v; p = i; }
        }
        __syncthreads();
        if (p != k) {
            float a  = M[k][t],        b  = M[p][t];
            float a2 = M[k][DDIM + t], b2 = M[p][DDIM + t];
            M[k][t] = b;  M[p][t] = a;
            M[k][DDIM + t] = b2; M[p][DDIM + t] = a2;
            det = -det;
        }
        __syncthreads();
        float piv = M[k][k];
        det *= piv;
        float rp = (fabsf(piv) > 1e-30f) ? (1.0f / piv) : 0.0f;
        M[k][t] *= rp;
        M[k][DDIM + t] *= rp;
        __syncthreads();
        if (t != k) {
            float f = M[t][k];
            for (int j = 0; j < 2 * DDIM; ++j) M[t][j] -= f * M[k][j];
        }
        __syncthreads();
    }

    for (int col = 0; col < DDIM; ++col) invSw[swz(t, col)] = M[t][DDIM + col];
    if (t == 0) *detOut = det;
}

// ---------------------------------------------------------------------------
// Kernel 2: main density kernel. 256 threads = 8 wave32's; each wave computes
// q for 16 samples via two chained V_WMMA_F32_16X16X4_F32 stages:
//   Y = D * invS   (64 WMMAs; Y stored *transposed* to a wave-private LDS
//                   tile with b128 stores)
//   Q = Y * D^T    (16 WMMAs), q = diag(Q)
// invS arrives pre-swizzled so every B-fragment is one contiguous ds_load_b64.
// ---------------------------------------------------------------------------
__global__ void __launch_bounds__(256)
mixture_density_kernel(const float* __restrict__ samples,
                       const float* __restrict__ Phi,
                       const float* __restrict__ mu,
                       const float* __restrict__ invSw_g,
                       const float* __restrict__ det_g,
                       float* __restrict__ out, int N) {
    __shared__ float sInv[DDIM * DDIM];          // 16 KB, fragment order
    __shared__ float sMu[DDIM];
    __shared__ float sYT[8][DDIM * 16];          // yT[col][row], 4 KB per wave

    const int tid = threadIdx.x;
    for (int i = tid; i < DDIM * DDIM; i += 256) sInv[i] = invSw_g[i];
    if (tid < DDIM) sMu[tid] = mu[tid];
    __syncthreads();

    const int wave = tid >> 5;
    const int lane = tid & 31;
    const int lo   = lane & 15;       // sample-in-tile (both half-waves)
    const int hi   = lane >> 4;       // which K sub-pair this lane carries

    const int n0 = (blockIdx.x * 8 + wave) * 16;   // first sample of the tile

    // Diff tile in fp32 WMMA A-layout (== B-layout of D^T, by symmetry):
    //   lane(lo,hi), chunk c: cols {4c+2hi, 4c+2hi+1} of row n0+lo.
    const float* rowp = samples + (size_t)(n0 + lo) * DDIM;
    v2f F[16];
#pragma unroll
    for (int c = 0; c < 16; ++c) {
        const int col = 4 * c + 2 * hi;
        v2f v = *(const v2f*)(rowp + col);
        v.x -= sMu[col];
        v.y -= sMu[col + 1];
        F[c] = v;
    }

    // Stage 1: Y(16x64) = D(16x64) @ invS(64x64), per 16-col output tile j.
    float* y = &sYT[wave][0];
#pragma unroll
    for (int j = 0; j < 4; ++j) {
        v8f acc = {};
#pragma unroll
        for (int c = 0; c < 16; ++c) {
            // Contiguous pre-swizzled B-fragment: single ds_load_b64.
            const int idx = ((c * 2 + hi) << 6) + (j << 4) + lo;
            v2f b = *(const v2f*)&sInv[2 * idx];
            acc = __builtin_amdgcn_wmma_f32_16x16x4_f32(
                false, F[c], false, b, (short)0, acc, false, false);
        }
        // Store C-layout result transposed: lane's 8 values are contiguous.
        //   yT[col = 16j+lo][rows 8hi .. 8hi+7]
        const int base = ((16 * j + lo) << 4) + 8 * hi;
        v4f lo4 = {acc[0], acc[1], acc[2], acc[3]};
        v4f hi4 = {acc[4], acc[5], acc[6], acc[7]};
        *(v4f*)&y[base]     = lo4;
        *(v4f*)&y[base + 4] = hi4;
    }

    // Stage 2: Q(16x16) = Y(16x64) @ D^T(64x16); B-fragment of D^T is F[c].
    // A-fragment of Y: {Y[lo][col], Y[lo][col+1]} = {yT[col][lo], yT[col+1][lo]}
    // -> one ds_load_2addr_b32 landing in a consecutive VGPR pair.
    v8f q = {};
#pragma unroll
    for (int c = 0; c < 16; ++c) {
        const int col = 4 * c + 2 * hi;
        v2f a;
        a.x = y[(col << 4) + lo];
        a.y = y[((col + 1) << 4) + lo];
        q = __builtin_amdgcn_wmma_f32_16x16x4_f32(
            false, a, false, F[c], (short)0, q, false, false);
    }

    // Extract diag(Q): row r lives at lane r (VGPR r); row r+8 at lane r+24.
    float qd = 0.0f;
#pragma unroll
    for (int r = 0; r < 8; ++r) {
        if (lane == r)      qd = q[r];
        if (lane == r + 24) qd = q[r];
    }

    const float det = *det_g;
    const float c0  = Phi[0] * rsqrtf(TWO_PI_F * det);
    const float res = -logf(c0 * expf(-0.5f * qd) + EPS_F);

    int row = -1;
    if (lane < 8)        row = n0 + lane;
    else if (lane >= 24) row = n0 + lane - 16;
    if (row >= 0 && row < N) out[row] = res;
}

// ---------------------------------------------------------------------------
// Scalar tail kernel (only launched if N is not a multiple of 128).
// Reads invSigma through the fragment-order index.
// ---------------------------------------------------------------------------
__global__ void mixture_tail_kernel(const float* __restrict__ samples,
                                    const float* __restrict__ Phi,
                                    const float* __restrict__ mu,
                                    const float* __restrict__ invSw,
                                    const float* __restrict__ det_g,
                                    float* __restrict__ out, int start, int N) {
    int n = start + blockIdx.x * blockDim.x + threadIdx.x;
    if (n >= N) return;
    const float* x = samples + (size_t)n * DDIM;
    float d[DDIM];
    for (int i = 0; i < DDIM; ++i) d[i] = x[i] - mu[i];
    float qv = 0.0f;
    for (int e = 0; e < DDIM; ++e) {
        float s = 0.0f;
        for (int k = 0; k < DDIM; ++k) s += d[k] * invSw[swz(k, e)];
        qv += s * d[e];
    }
    float c0 = Phi[0] * rsqrtf(TWO_PI_F * (*det_g));
    out[n] = -logf(c0 * expf(-0.5f * qv) + EPS_F);
}

extern "C" void kernel_launch(void* const* d_in, const int* in_sizes, int n_in,
                              void* d_out, int out_size, void* d_ws, size_t ws_size,
                              hipStream_t stream) {
    const float* samples = (const float*)d_in[0];   // [N, 64]
    const float* Phi     = (const float*)d_in[1];   // [1]
    const float* mu      = (const float*)d_in[2];   // [64]
    const float* Sigma   = (const float*)d_in[3];   // [64, 64]
    float* out = (float*)d_out;

    const int N = in_sizes[0] / DDIM;

    float* invSw = (float*)d_ws;           // 4096 floats, fragment order
    float* det   = invSw + DDIM * DDIM;    // 1 float

    invdet64_kernel<<<dim3(1), dim3(DDIM), 0, stream>>>(Sigma, invSw, det);

    const int nTiles  = N / 16;
    const int nBlocks = nTiles / 8;                 // 128 samples per block
    const int covered = nBlocks * 128;
    if (nBlocks > 0)
        mixture_density_kernel<<<dim3(nBlocks), dim3(256), 0, stream>>>(
            samples, Phi, mu, invSw, det, out, N);
    if (covered < N) {
        const int tail = N - covered;
        mixture_tail_kernel<<<dim3((tail + 255) / 256), dim3(256), 0, stream>>>(
            samples, Phi, mu, invSw, det, out, covered, N);
    }
}